// Sampler_90804198572083
// MI455X (gfx1250) — compile-verified
//
#include <hip/hip_runtime.h>
#include <math.h>

// Sampler: B=128 rows, V=128000 vocab. One workgroup (1024 thr = 32 wave32
// waves) per row. Sorting replaced by radix-select over the monotone
// float-order key of the logit (softmax is monotone => prob order == logit
// order; every filter is a key threshold).
//
// Row passes (all float4/B128; the 65MB input stays resident in the 192MB L2):
//   max (1), dual top-p/top-k radix select (4; level-0 mass total == Zs, so no
//   separate sum-exp pass), sampling radix select (4; level-0 kept-mass total
//   == totE, so no separate total pass), fused logprob-write + sampled-index
//   (1)  => 10 full-row passes.
// HBM floor: read 65MB + write 65MB ~ 5.6us @ 23.3 TB/s.

#define BB    128
#define VV    128000
#define NV4   (VV / 4)
#define NTHR  1024
#define NWAVE (NTHR / 32)
#define NH    8          // LDS histogram replicas (contention relief)

#define NEG_FLT_MAX (-3.40282347e+38f)   // jnp.finfo(float32).min

__device__ __forceinline__ unsigned fkey(float f) {
    unsigned u = __float_as_uint(f);
    return (u & 0x80000000u) ? ~u : (u | 0x80000000u);   // bigger float -> bigger key
}

__device__ __forceinline__ float waveMax(float v) {
    #pragma unroll
    for (int o = 16; o > 0; o >>= 1) v = fmaxf(v, __shfl_down(v, o, 32));
    return v;
}
__device__ __forceinline__ float blockMax(float v, float* partial) {
    int lane = threadIdx.x & 31, wid = threadIdx.x >> 5;
    v = waveMax(v);
    if (lane == 0) partial[wid] = v;
    __syncthreads();
    if (wid == 0) {
        float r = (lane < NWAVE) ? partial[lane] : NEG_FLT_MAX;
        r = waveMax(r);
        if (lane == 0) partial[0] = r;
    }
    __syncthreads();
    float r = partial[0];
    __syncthreads();
    return r;
}

struct SelOut {
    unsigned selB;    // selected bucket
    float    r2;      // residual threshold inside the selected bucket
    float    sSel;    // mass/count of the selected bucket
    float    above;   // mass/count in buckets strictly above the selection
    float    total;   // total over all 256 buckets
};

// Wave-cooperative descending crossing search over 256 buckets (one wave32).
// Lane j owns buckets 255-8j..248-8j; __shfl_up exclusive scan gives per-lane
// prefixes; __ballot picks the highest-bucket crossing.  If frac >= 0 the
// threshold is frac*total (lets level 0 derive Zs / totE from its own
// histogram); otherwise residDirect.  strict selects '>' vs '>=' crossing.
// Fallback (no crossing): descend into the lowest populated bucket, which
// makes the final key the smallest present key (i.e. "keep everything").
__device__ void waveCross(const float* b256, float residDirect, float frac,
                          bool strict, SelOut* out)
{
    const int lane = threadIdx.x & 31;
    const int bhi  = 255 - 8 * lane;
    float Lj = 0.0f;
    #pragma unroll
    for (int q = 0; q < 8; ++q) Lj += b256[bhi - q];
    float run = Lj;
    #pragma unroll
    for (int o = 1; o < 32; o <<= 1) {
        float t = __shfl_up(run, o, 32);
        if (lane >= o) run += t;
    }
    const float total = __shfl(run, 31, 32);
    const float resid = (frac >= 0.0f) ? frac * total : residDirect;
    float acc = run - Lj;                                   // exclusive prefix
    int   selB = -1;
    float r2 = 0.0f, sSel = 0.0f;
    #pragma unroll
    for (int q = 0; q < 8; ++q) {
        const int   bb = bhi - q;
        const float s  = b256[bb];
        const bool  cross = strict ? (acc + s > resid) : (acc + s >= resid);
        if (selB < 0 && s > 0.0f && cross) { selB = bb; r2 = resid - acc; sSel = s; }
        acc += s;
    }
    unsigned long long bal = __ballot(selB >= 0);
    if (bal != 0ull) {
        int win = __ffsll(bal) - 1;                         // lowest lane = highest bucket
        if (lane == win) {
            out->selB = (unsigned)selB; out->r2 = r2; out->sSel = sSel;
            out->above = resid - r2;    out->total = total;
        }
    } else if (lane == 0) {                                 // no crossing: keep-all path
        int sel = 0;
        for (int bb = 0; bb < 256; ++bb)
            if (b256[bb] > 0.0f) { sel = bb; break; }
        const float s = b256[sel];
        out->selB = (unsigned)sel; out->r2 = s; out->sSel = s;
        out->above = total - s;    out->total = total;
    }
}

__global__ __launch_bounds__(NTHR)
void Sampler_90804198572083_kernel(const float* __restrict__ logits,
                                   const float* __restrict__ temps,
                                   const int*   __restrict__ topks,
                                   const float* __restrict__ topps,
                                   const float* __restrict__ minps,
                                   const float* __restrict__ us,
                                   float* __restrict__ out)
{
    __shared__ float    histS[NH][256];     // mass histogram
    __shared__ float    histC[NH][256];     // count histogram (dual select only)
    __shared__ float    partial[NWAVE];
    __shared__ SelOut   soA, soB;
    __shared__ unsigned s_selIdx;

    const int row = blockIdx.x;
    const int tid = threadIdx.x;
    const int wid = tid >> 5;
    const float*  __restrict__ rp  = logits + (size_t)row * VV;
    const float4* __restrict__ rp4 = reinterpret_cast<const float4*>(rp);

    const float T    = temps[row];
    const float invT = 1.0f / T;
    const float a    = 1.4426950408889634f * invT;  // log2(e)/T ; e_i = exp2((l-Lmax)*a)
    const int   topk = topks[row];
    const float topp = topps[row];
    const float minp = minps[row];
    const float u    = us[row];

    // ---- pass 1: row max ----
    float m = NEG_FLT_MAX;
    for (int i4 = tid; i4 < NV4; i4 += NTHR) {
        if (i4 + 4 * NTHR < NV4) __builtin_prefetch(rp4 + i4 + 4 * NTHR, 0, 0); // global_prefetch_b8
        float4 v = rp4[i4];
        m = fmaxf(fmaxf(m, fmaxf(v.x, v.y)), fmaxf(v.z, v.w));
    }
    const float Lmax = blockMax(m, partial);

    // ---- passes 2-5: dual radix select -------------------------------------
    //  top-p: mass histogram, threshold topp*Zs (Zs = level-0 total), strict >
    //  top-k: count histogram, threshold topk, crossing >=
    // wave 0 searches the mass buckets, wave 1 the count buckets, in parallel.
    unsigned prefP = 0, prefK = 0;
    float residP = 0.0f, residK = (float)topk;
    float accAboveP = 0.0f, lastSelP = 0.0f;
    for (int level = 0; level < 4; ++level) {
        const int shift = 24 - 8 * level;
        for (int b = tid; b < NH * 256; b += NTHR) {
            ((float*)histS)[b] = 0.0f;
            ((float*)histC)[b] = 0.0f;
        }
        __syncthreads();
        const unsigned hi_mask = (level == 0) ? 0u : (0xFFFFFFFFu << (shift + 8));
        const int h = (tid >> 7) & (NH - 1);

        for (int i4 = tid; i4 < NV4; i4 += NTHR) {
            float4 v = rp4[i4];
            float ls[4] = {v.x, v.y, v.z, v.w};
            #pragma unroll
            for (int c = 0; c < 4; ++c) {
                const float    l   = ls[c];
                const unsigned k   = fkey(l);
                const unsigned bkt = (k >> shift) & 255;
                if (level == 0 || (k & hi_mask) == (prefP & hi_mask))
                    atomicAdd(&histS[h][bkt], exp2f((l - Lmax) * a));  // ds_add_f32
                if (level == 0 || (k & hi_mask) == (prefK & hi_mask))
                    atomicAdd(&histC[h][bkt], 1.0f);
            }
        }
        __syncthreads();
        for (int b = tid; b < 256; b += NTHR) {              // fold replicas
            float s = 0.0f, c = 0.0f;
            #pragma unroll
            for (int r = 0; r < NH; ++r) { s += histS[r][b]; c += histC[r][b]; }
            histS[0][b] = s; histC[0][b] = c;
        }
        __syncthreads();
        if (wid == 0) waveCross(histS[0], residP, (level == 0) ? topp : -1.0f, true,  &soA);
        if (wid == 1) waveCross(histC[0], residK, -1.0f,                       false, &soB);
        __syncthreads();
        prefP |= soA.selB << shift;  residP = soA.r2;
        accAboveP += soA.above;      lastSelP = soA.sSel;
        prefK |= soB.selB << shift;  residK = soB.r2;
        __syncthreads();
    }
    const unsigned tP = prefP, tK = prefK;
    const float    normE   = accAboveP + lastSelP;           // mass with key >= tP
    const float    logNorm = logf(normE);
    const unsigned tS = (tP > tK) ? tP : tK;

    // ---- passes 6-9: inverse-CDF radix select over kept set ----------------
    // kept = key >= tS  &&  e >= min_p  (e of the max token is exactly 1, so
    // thr = p_max*min_p reduces to e >= min_p). Level-0 total == totE and the
    // draw target is u*totE, computed inside the level-0 crossing search.
    unsigned prefX = 0;
    float    residX = 0.0f;
    for (int level = 0; level < 4; ++level) {
        const int shift = 24 - 8 * level;
        for (int b = tid; b < NH * 256; b += NTHR) ((float*)histS)[b] = 0.0f;
        __syncthreads();
        const unsigned hi_mask = (level == 0) ? 0u : (0xFFFFFFFFu << (shift + 8));
        const int h = (tid >> 7) & (NH - 1);

        for (int i4 = tid; i4 < NV4; i4 += NTHR) {
            float4 v = rp4[i4];
            float ls[4] = {v.x, v.y, v.z, v.w};
            #pragma unroll
            for (int c = 0; c < 4; ++c) {
                const float    l = ls[c];
                const unsigned k = fkey(l);
                if (k < tS) continue;
                if (level != 0 && (k & hi_mask) != (prefX & hi_mask)) continue;
                const float e = exp2f((l - Lmax) * a);
                if (e < minp) continue;
                atomicAdd(&histS[h][(k >> shift) & 255], e);
            }
        }
        __syncthreads();
        for (int b = tid; b < 256; b += NTHR) {
            float s = 0.0f;
            #pragma unroll
            for (int r = 0; r < NH; ++r) s += histS[r][b];
            histS[0][b] = s;
        }
        __syncthreads();
        if (wid == 0) waveCross(histS[0], residX, (level == 0) ? u : -1.0f, false, &soA);
        __syncthreads();
        prefX |= soA.selB << shift;  residX = soA.r2;
        __syncthreads();
    }
    const unsigned tX = prefX;

    // ---- pass 10: fused logprob write + sampled-token index (min idx) ------
    if (tid == 0) s_selIdx = 0xFFFFFFFFu;
    __syncthreads();
    float4* __restrict__ lpOut4 =
        reinterpret_cast<float4*>(out + BB + (size_t)row * VV);
    for (int i4 = tid; i4 < NV4; i4 += NTHR) {
        float4 v = rp4[i4];
        float ls[4] = {v.x, v.y, v.z, v.w};
        float lp[4];
        #pragma unroll
        for (int c = 0; c < 4; ++c) {
            const float    l = ls[c];
            const unsigned k = fkey(l);
            lp[c] = (k >= tP) ? ((l - Lmax) * invT - logNorm) : NEG_FLT_MAX;
            if (k == tX && k >= tS) {
                const float e = exp2f((l - Lmax) * a);
                if (e >= minp) atomicMin(&s_selIdx, (unsigned)(4 * i4 + c)); // ds_min_u32
            }
        }
        lpOut4[i4] = make_float4(lp[0], lp[1], lp[2], lp[3]);
    }
    __syncthreads();

    if (tid == 0) {
        unsigned selIdx = s_selIdx;
        if (selIdx == 0xFFFFFFFFu) selIdx = 0;               // unreachable safety
        out[row] = (float)selIdx;                            // token_ids
        const float lsel = rp[selIdx];
        out[BB + (size_t)BB * VV + row] = (lsel - Lmax) * invT - logNorm; // next_token_logprobs
    }
}

extern "C" void kernel_launch(void* const* d_in, const int* in_sizes, int n_in,
                              void* d_out, int out_size, void* d_ws, size_t ws_size,
                              hipStream_t stream) {
    const float* logits = (const float*)d_in[0];
    const float* temps  = (const float*)d_in[1];
    const int*   topks  = (const int*)  d_in[2];
    const float* topps  = (const float*)d_in[3];
    const float* minps  = (const float*)d_in[4];
    const float* us     = (const float*)d_in[5];
    (void)in_sizes; (void)n_in; (void)out_size; (void)d_ws; (void)ws_size;

    Sampler_90804198572083_kernel<<<BB, NTHR, 0, stream>>>(
        logits, temps, topks, topps, minps, us, (float*)d_out);
}